// LearnableWeightedRGCN_20564303413375
// MI455X (gfx1250) — compile-verified
//
#include <hip/hip_runtime.h>
#include <math.h>

// ---------------------------------------------------------------------------
// LearnableWeightedRGCN for MI455X (gfx1250, wave32).
// Dense GEMMs use V_WMMA_F32_16X16X4_F32 (exact fp32 tensor op) since the
// SpMM (random 1KB gathers + 820M f32 atomics) dominates the roofline, not
// the 39 GFLOP of dense math.
// ---------------------------------------------------------------------------

#define N_NODES 50000
#define DIM     256
#define NREL    4
#define NEDGE   800000
#define LN_EPS  1e-5f
#define NEG_INF -1e30f

typedef __attribute__((ext_vector_type(2))) float v2f;
typedef __attribute__((ext_vector_type(8))) float v8f;

__device__ __forceinline__ float gelu_erf(float v) {
  // exact erf GELU (torch / jax approximate=False)
  return 0.5f * v * (1.0f + erff(v * 0.70710678118654752440f));
}

// ---------------------------------------------------------------------------
// C[N,256] = A[N,256] @ B[256,256] (+ bias).  fp32 WMMA 16x16x4.
// Block = 256 threads (8 waves) = one 16-row panel of A staged in LDS.
// Wave w computes columns [w*32, w*32+32) as two independent 16x16 tiles.
// ---------------------------------------------------------------------------
__global__ __launch_bounds__(256) void gemm256_kernel(const float* __restrict__ A,
                                                      const float* __restrict__ B,
                                                      const float* __restrict__ bias,
                                                      float* __restrict__ C) {
  __shared__ float sA[16 * 260];  // pad 256->260 so lanes hit distinct banks
  const int tid = threadIdx.x;
  const int wave = tid >> 5;
  const int lane = tid & 31;
  const int blockRow = blockIdx.x * 16;

  // Cooperative stage of the 16x256 A panel (coalesced: 256 lanes per row).
#pragma unroll
  for (int i = 0; i < 16; ++i)
    sA[i * 260 + tid] = A[(size_t)(blockRow + i) * DIM + tid];
  __syncthreads();

  const int lh = lane >> 4;   // lane half: selects K pair per ISA A layout
  const int ll = lane & 15;   // M (for A) / N (for B) index within tile
  const int n0 = wave * 32;   // this wave's 32-column group

  v8f acc0 = {};
  v8f acc1 = {};
  for (int k0 = 0; k0 < DIM; k0 += 4) {
    const int kb = k0 + lh * 2;  // VGPR0 holds K=kb, VGPR1 holds K=kb+1
    v2f a, b0, b1;
    a.x  = sA[ll * 260 + kb];
    a.y  = sA[ll * 260 + kb + 1];
    b0.x = B[(size_t)kb * DIM + n0 + ll];
    b0.y = B[(size_t)(kb + 1) * DIM + n0 + ll];
    b1.x = B[(size_t)kb * DIM + n0 + 16 + ll];
    b1.y = B[(size_t)(kb + 1) * DIM + n0 + 16 + ll];
    // D = A*B + C : two independent accumulators -> co-executable WMMAs
    acc0 = __builtin_amdgcn_wmma_f32_16x16x4_f32(false, a, false, b0, (short)0, acc0, false, false);
    acc1 = __builtin_amdgcn_wmma_f32_16x16x4_f32(false, a, false, b1, (short)0, acc1, false, false);
  }

  // C/D layout: VGPR v, lanes 0-15 -> M=v, lanes 16-31 -> M=v+8; N = lane&15.
  const int col0 = n0 + ll;
  const int col1 = n0 + 16 + ll;
  const float bv0 = bias ? bias[col0] : 0.0f;
  const float bv1 = bias ? bias[col1] : 0.0f;
#pragma unroll
  for (int v = 0; v < 8; ++v) {
    const int m = blockRow + v + lh * 8;
    C[(size_t)m * DIM + col0] = acc0[v] + bv0;
    C[(size_t)m * DIM + col1] = acc1[v] + bv1;
  }
}

// ---------------------------------------------------------------------------
// x = GELU(LayerNorm(in))   (row-wise, wave per row, 8 features per lane)
// ---------------------------------------------------------------------------
__global__ __launch_bounds__(256) void ln_gelu_kernel(const float* __restrict__ in,
                                                      const float* __restrict__ g,
                                                      const float* __restrict__ b,
                                                      float* __restrict__ out) {
  const int wave = threadIdx.x >> 5;
  const int lane = threadIdx.x & 31;
  const int row = blockIdx.x * 8 + wave;
  if (row >= N_NODES) return;

  float v[8];
  float s = 0.f;
#pragma unroll
  for (int j = 0; j < 8; ++j) {
    v[j] = in[(size_t)row * DIM + lane + j * 32];
    s += v[j];
  }
#pragma unroll
  for (int off = 16; off > 0; off >>= 1) s += __shfl_xor(s, off, 32);
  const float mean = s * (1.0f / DIM);

  float q = 0.f;
#pragma unroll
  for (int j = 0; j < 8; ++j) { const float d = v[j] - mean; q += d * d; }
#pragma unroll
  for (int off = 16; off > 0; off >>= 1) q += __shfl_xor(q, off, 32);
  const float rstd = rsqrtf(q * (1.0f / DIM) + LN_EPS);

#pragma unroll
  for (int j = 0; j < 8; ++j) {
    const int c = lane + j * 32;
    const float y = (v[j] - mean) * rstd * g[c] + b[c];
    out[(size_t)row * DIM + c] = gelu_erf(y);
  }
}

// ---------------------------------------------------------------------------
// SpMM scatter: msgs[row] += val * xrel[col]   (wave per edge)
// ---------------------------------------------------------------------------
__global__ __launch_bounds__(256) void spmm_kernel(const int* __restrict__ rows,
                                                   const int* __restrict__ cols,
                                                   const float* __restrict__ vals,
                                                   const float* __restrict__ xrel,
                                                   float* __restrict__ msgs) {
  const long long gid = (long long)blockIdx.x * blockDim.x + threadIdx.x;
  const long long e = gid >> 5;
  const int lane = (int)(gid & 31);
  if (e >= NEDGE) return;
  const int r = rows[e];
  const int c = cols[e];
  const float w = vals[e];
  const float* __restrict__ src = xrel + (size_t)c * DIM;
  float* __restrict__ dst = msgs + (size_t)r * DIM;
#pragma unroll
  for (int j = 0; j < 8; ++j) {
    const int d = lane + j * 32;  // lane-contiguous: coalesced 128B gathers
    atomicAdd(&dst[d], w * src[d]);
  }
}

// ---------------------------------------------------------------------------
// Per-node: semantic attention over R relations, + bias + self path,
// GELU, final LayerNorm.  Wave per node.
// ---------------------------------------------------------------------------
__global__ __launch_bounds__(256) void finalize_kernel(const float* __restrict__ msgs,     // [R][N][D]
                                                       const float* __restrict__ selfout,  // [N][D]
                                                       const float* __restrict__ attn_vec, // [D]
                                                       const float* __restrict__ bias,     // [D]
                                                       const float* __restrict__ ln_g,
                                                       const float* __restrict__ ln_b,
                                                       float* __restrict__ final_out,      // [N][D]
                                                       float* __restrict__ attn_out) {     // [N][R]
  const int wave = threadIdx.x >> 5;
  const int lane = threadIdx.x & 31;
  const int n = blockIdx.x * 8 + wave;
  if (n >= N_NODES) return;

  float m[NREL][8];
  float score[NREL];
  float nzf[NREL];
#pragma unroll
  for (int r = 0; r < NREL; ++r) {
    float sc = 0.f;
    float anynz = 0.f;
#pragma unroll
    for (int j = 0; j < 8; ++j) {
      const int d = lane + j * 32;
      const float x = msgs[((size_t)r * N_NODES + n) * DIM + d];
      m[r][j] = x;
      sc += x * attn_vec[d];
      if (x != 0.f) anynz = 1.f;
    }
#pragma unroll
    for (int off = 16; off > 0; off >>= 1) {
      sc += __shfl_xor(sc, off, 32);
      anynz += __shfl_xor(anynz, off, 32);
    }
    nzf[r] = (anynz > 0.f) ? 1.f : 0.f;
    score[r] = (anynz > 0.f) ? sc : NEG_INF;
  }

  // softmax over relations (uniform across lanes), then mask
  float mx = score[0];
#pragma unroll
  for (int r = 1; r < NREL; ++r) mx = fmaxf(mx, score[r]);
  float a[NREL];
  float den = 0.f;
#pragma unroll
  for (int r = 0; r < NREL; ++r) { a[r] = expf(score[r] - mx); den += a[r]; }
  const float inv = 1.0f / den;
#pragma unroll
  for (int r = 0; r < NREL; ++r) a[r] = a[r] * inv * nzf[r];
  if (lane == 0) {
#pragma unroll
    for (int r = 0; r < NREL; ++r) attn_out[(size_t)n * NREL + r] = a[r];
  }

  // out = attn-weighted msgs + bias + self path, then GELU
  float o[8];
  float s = 0.f;
#pragma unroll
  for (int j = 0; j < 8; ++j) {
    const int d = lane + j * 32;
    float acc = bias[d] + selfout[(size_t)n * DIM + d];
#pragma unroll
    for (int r = 0; r < NREL; ++r) acc += a[r] * m[r][j];
    o[j] = gelu_erf(acc);
    s += o[j];
  }

  // final LayerNorm
#pragma unroll
  for (int off = 16; off > 0; off >>= 1) s += __shfl_xor(s, off, 32);
  const float mean = s * (1.0f / DIM);
  float q = 0.f;
#pragma unroll
  for (int j = 0; j < 8; ++j) { const float d = o[j] - mean; q += d * d; }
#pragma unroll
  for (int off = 16; off > 0; off >>= 1) q += __shfl_xor(q, off, 32);
  const float rstd = rsqrtf(q * (1.0f / DIM) + LN_EPS);

#pragma unroll
  for (int j = 0; j < 8; ++j) {
    const int c = lane + j * 32;
    final_out[(size_t)n * DIM + c] = (o[j] - mean) * rstd * ln_g[c] + ln_b[c];
  }
}

// ---------------------------------------------------------------------------
extern "C" void kernel_launch(void* const* d_in, const int* in_sizes, int n_in,
                              void* d_out, int out_size, void* d_ws, size_t ws_size,
                              hipStream_t stream) {
  (void)in_sizes; (void)n_in; (void)out_size; (void)ws_size;

  const int*   rows     = (const int*)d_in[0];    // [R,E]
  const int*   cols     = (const int*)d_in[1];    // [R,E]
  const float* vals     = (const float*)d_in[2];  // [R,E]
  const float* node_emb = (const float*)d_in[3];  // [N,D]
  const float* pre_W    = (const float*)d_in[4];  // [D,D]
  const float* pre_b    = (const float*)d_in[5];  // [D]
  const float* ln1_g    = (const float*)d_in[6];
  const float* ln1_b    = (const float*)d_in[7];
  const float* rel_W    = (const float*)d_in[8];  // [R,D,D]
  const float* attn_vec = (const float*)d_in[9];  // [D]
  const float* self_W   = (const float*)d_in[10]; // [D,D]
  const float* self_b   = (const float*)d_in[11];
  const float* bias     = (const float*)d_in[12];
  const float* ln2_g    = (const float*)d_in[13];
  const float* ln2_b    = (const float*)d_in[14];

  // workspace layout (floats): x | tmp(->selfout) | xrel(single rel) | msgs[R]
  float* ws   = (float*)d_ws;
  float* x    = ws;
  float* tmp  = x    + (size_t)N_NODES * DIM;
  float* xrel = tmp  + (size_t)N_NODES * DIM;
  float* msgs = xrel + (size_t)N_NODES * DIM;

  // output layout: final [N*D] | attn [N*R] | x_all [N*D]
  float* final_out = (float*)d_out;
  float* attn_out  = final_out + (size_t)N_NODES * DIM;
  float* xall_out  = attn_out  + (size_t)N_NODES * NREL;

  const dim3 blk(256);
  const int gemm_grid = N_NODES / 16;       // 3125 (exact)
  const int row_grid  = N_NODES / 8;        // 6250 (exact)
  const int spmm_grid = (NEDGE * 32) / 256; // 100000 (exact)

  // 1) tmp = node_emb @ pre_W + pre_b
  gemm256_kernel<<<gemm_grid, blk, 0, stream>>>(node_emb, pre_W, pre_b, tmp);
  // 2) x = GELU(LN1(tmp))
  ln_gelu_kernel<<<row_grid, blk, 0, stream>>>(tmp, ln1_g, ln1_b, x);
  // 3) zero message planes
  hipMemsetAsync(msgs, 0, (size_t)NREL * N_NODES * DIM * sizeof(float), stream);
  // 4) per relation: xrel = x @ rel_W[r]; msgs[r] += scatter(vals * xrel[cols])
  for (int r = 0; r < NREL; ++r) {
    gemm256_kernel<<<gemm_grid, blk, 0, stream>>>(x, rel_W + (size_t)r * DIM * DIM, nullptr, xrel);
    spmm_kernel<<<spmm_grid, blk, 0, stream>>>(rows + (size_t)r * NEDGE,
                                               cols + (size_t)r * NEDGE,
                                               vals + (size_t)r * NEDGE,
                                               xrel,
                                               msgs + (size_t)r * N_NODES * DIM);
  }
  // 5) selfout = x @ self_W + self_b  (reuse tmp)
  gemm256_kernel<<<gemm_grid, blk, 0, stream>>>(x, self_W, self_b, tmp);
  // 6) attention + combine + GELU + LN2
  finalize_kernel<<<row_grid, blk, 0, stream>>>(msgs, tmp, attn_vec, bias, ln2_g, ln2_b,
                                                final_out, attn_out);
  // 7) x_all output is just the embedding table
  hipMemcpyAsync(xall_out, node_emb, (size_t)N_NODES * DIM * sizeof(float),
                 hipMemcpyDeviceToDevice, stream);
}